// Propagate_33208687133414
// MI455X (gfx1250) — compile-verified
//
#include <hip/hip_runtime.h>
#include <stdint.h>

// CSR SpMM: out[i,:] = sum_{k in [indptr[i], indptr[i+1])} values[k] * x[indices[k],:]
// N=100K rows, E=1.6M edges, D=64. Bandwidth-bound gather (AI ~0.5 FLOP/B); x (25.6MB)
// is L2-resident on MI455X (192MB L2), so WMMA is irrelevant — the win is data movement.
//
// CDNA5 path (preferred): TDM gather mode — one tensor_load_to_lds with 8x 32-bit row
// indices pulls 8 arbitrary 256B x-rows into LDS per instruction (TENSORcnt,
// s_wait_tensorcnt). Double-buffered: batch i+1's gather DMA overlaps batch i's FMAs.
// Fallbacks: per-edge global_load_async_to_lds_b64 (ASYNCcnt) -> direct VGPR gather.

#define DFEAT 64
#define WAVES_PER_BLOCK 8
#define THREADS_PER_BLOCK (WAVES_PER_BLOCK * 32)
#define BATCH 8

#if defined(__has_builtin)
#  if __has_builtin(__builtin_amdgcn_tensor_load_to_lds)
#    define USE_TDM 1
#  endif
#  if !defined(USE_TDM) && __has_builtin(__builtin_amdgcn_global_load_async_to_lds_b64)
#    define USE_ASYNC 1
#  endif
#endif
#ifndef USE_TDM
#define USE_TDM 0
#endif
#ifndef USE_ASYNC
#define USE_ASYNC 0
#endif

#define AS1 __attribute__((address_space(1)))
#define AS3 __attribute__((address_space(3)))

typedef int          v2i   __attribute__((vector_size(8)));
typedef float        f32x2 __attribute__((vector_size(8)));
typedef unsigned int u32x4 __attribute__((vector_size(16)));
typedef int          i32x4 __attribute__((vector_size(16)));
typedef int          i32x8 __attribute__((vector_size(32)));

// Real compiler barriers + guaranteed mnemonics on the gfx1250 assembler.
#define WAIT_DS_0() asm volatile("s_wait_dscnt 0x0" ::: "memory")
#if USE_TDM
// One TDM op per batch: 2 outstanding -> wait<=1 retires the current batch.
#define WAIT_CUR_FULL() asm volatile("s_wait_tensorcnt 0x1" ::: "memory")
#define WAIT_ALL()      asm volatile("s_wait_tensorcnt 0x0" ::: "memory")
#else
// 8 async ops per full batch: wait<=8 retires the current batch (in-order).
#define WAIT_CUR_FULL() asm volatile("s_wait_asynccnt 0x8" ::: "memory")
#define WAIT_ALL()      asm volatile("s_wait_asynccnt 0x0" ::: "memory")
#endif

__global__ __launch_bounds__(THREADS_PER_BLOCK) void spmm_csr_tdm(
    const float* __restrict__ x,           // [N, 64]
    const long long* __restrict__ indptr,  // [N+1]
    const long long* __restrict__ indices, // [E]
    const float* __restrict__ values,      // [E]
    float* __restrict__ out,               // [N, 64]
    int n_rows)
{
#if USE_TDM || USE_ASYNC
    __shared__ float sbuf[WAVES_PER_BLOCK][2][BATCH][DFEAT];  // 32 KB / block
#endif

    const int wave = threadIdx.x >> 5;
    const int lane = threadIdx.x & 31;
    const int row  = blockIdx.x * WAVES_PER_BLOCK + wave;
    if (row >= n_rows) return;  // wave-uniform: EXEC stays all-ones inside

    // E < 2^31, N*32 < 2^31: 32-bit offsets keep gathers in SADDR+voffset form.
    const int start = (int)indptr[row];
    const int end   = (int)indptr[row + 1];

    float accx = 0.0f, accy = 0.0f;

#if USE_TDM || USE_ASYNC
    float vA[BATCH], vB[BATCH];

    // Issue one batch of up to 8 gathered 256B x-rows into LDS half `half`.
    auto issue_batch = [&](int half, int ebase, int m, float* v) {
        // WAR fence: prior consume's ds_loads of this half must complete before the
        // DMA engine may overwrite it (async/TDM LDS writes are unordered w.r.t. DS).
        WAIT_DS_0();
#if USE_TDM
        int idx[BATCH];
#pragma unroll
        for (int k = 0; k < BATCH; ++k) {
            if (k < m) {
                idx[k] = __builtin_amdgcn_readfirstlane((int)indices[ebase + k]);
                v[k]   = values[ebase + k];
            } else {
                idx[k] = 0;
            }
        }
        const unsigned lds_addr = (unsigned)__builtin_amdgcn_readfirstlane(
            (int)(uintptr_t)(AS3 void*)&sbuf[wave][half][0][0]);
        const int mm = __builtin_amdgcn_readfirstlane(m);
        const unsigned long long ga = (unsigned long long)(uintptr_t)x;
        const unsigned nr = (unsigned)n_rows;
        // D# group0: count=1 | gather_index_size=32b (bit30) | gather_mode (bit31);
        //            lds_addr; global_addr[56:0] of tensor base; type=2 (bits127:126).
        u32x4 g0 = { 0xC0000001u,
                     lds_addr,
                     (unsigned)(ga & 0xFFFFFFFFu),
                     (unsigned)((ga >> 32) & 0x01FFFFFFu) | 0x80000000u };
        // D# group1: workgroup_mask=0, data_size=2 (4B); tensor_dim0=64 (bits 79:48);
        //            tensor_dim1=n_rows (bits 111:80); tile_dim0=64 (bits 127:112);
        //            tile_dim1=m (#indices, bits 143:128); tensor_dim0_stride=64.
        i32x8 g1 = { (int)0x00020000u,
                     (int)(64u << 16),
                     (int)((nr & 0xFFFFu) << 16),
                     (int)(((nr >> 16) & 0xFFFFu) | (64u << 16)),
                     mm,
                     64, 0, 0 };
        // D# groups 2-3: 32-bit row indices 0..7.
        i32x4 g2 = { idx[0], idx[1], idx[2], idx[3] };
        i32x4 g3 = { idx[4], idx[5], idx[6], idx[7] };
        // clang-23 6-arg form: trailing int32x8 = VADDR4, "unused (set to NULL)".
        i32x8 g4 = { 0, 0, 0, 0, 0, 0, 0, 0 };
        __builtin_amdgcn_tensor_load_to_lds(g0, g1, g2, g3, g4, /*cpol=*/0);
#else
#pragma unroll
        for (int k = 0; k < BATCH; ++k) {
            if (k < m) {
                const int c = (int)indices[ebase + k];
                v[k] = values[ebase + k];
                float* gsrc = const_cast<float*>(x) + (size_t)c * DFEAT + 2 * lane;
                __builtin_amdgcn_global_load_async_to_lds_b64(
                    (AS1 v2i*)gsrc,
                    (AS3 v2i*)&sbuf[wave][half][k][2 * lane],
                    /*offset=*/0, /*cpol=*/0);
            }
        }
#endif
    };

    auto consume_batch = [&](int half, int m, const float* v) {
        // Launder the LDS pointer: the TDM builtin carries no pointer, so the
        // compiler cannot see sbuf being written — stop it folding these ds_loads.
        const AS3 float* p = (const AS3 float*)&sbuf[wave][half][0][0];
        asm volatile("" : "+v"(p));
#pragma unroll
        for (int k = 0; k < BATCH; ++k) {
            if (k < m) {
                const f32x2 a = *(const AS3 f32x2*)(p + k * DFEAT + 2 * lane);
                accx = fmaf(v[k], a[0], accx);
                accy = fmaf(v[k], a[1], accy);
            }
        }
    };

    const int cnt = end - start;
    int m_cur = cnt < BATCH ? cnt : BATCH;
    int h = 0;
    if (m_cur > 0) issue_batch(0, start, m_cur, vA);
    int e = start + m_cur;
    while (m_cur > 0) {
        int m_next = end - e;
        if (m_next > BATCH) m_next = BATCH;
        if (m_next > 0) issue_batch(h ^ 1, e, m_next, (h == 0) ? vB : vA);
#if USE_TDM
        if (m_next > 0) { WAIT_CUR_FULL(); } else { WAIT_ALL(); }
#else
        if (m_next == BATCH) { WAIT_CUR_FULL(); } else { WAIT_ALL(); }
#endif
        consume_batch(h, m_cur, (h == 0) ? vA : vB);
        e += m_next;
        m_cur = m_next;
        h ^= 1;
    }
#else
    // Fallback: direct-gather, 4 independent 256B gathers in flight per wave.
    const float2* __restrict__ x2 = reinterpret_cast<const float2*>(x);
    int e = start;
    for (; e + 3 < end; e += 4) {
        const int ep = e + 8;
        if (ep < end) {
            const int pcol = (int)indices[ep];
            __builtin_prefetch(&x[(long long)pcol * DFEAT], 0, 0);
        }
        const int c0 = (int)indices[e + 0];
        const int c1 = (int)indices[e + 1];
        const int c2 = (int)indices[e + 2];
        const int c3 = (int)indices[e + 3];
        const float v0 = values[e + 0];
        const float v1 = values[e + 1];
        const float v2 = values[e + 2];
        const float v3 = values[e + 3];
        const float2 a0 = x2[c0 * (DFEAT / 2) + lane];
        const float2 a1 = x2[c1 * (DFEAT / 2) + lane];
        const float2 a2 = x2[c2 * (DFEAT / 2) + lane];
        const float2 a3 = x2[c3 * (DFEAT / 2) + lane];
        accx = fmaf(v0, a0.x, accx); accy = fmaf(v0, a0.y, accy);
        accx = fmaf(v1, a1.x, accx); accy = fmaf(v1, a1.y, accy);
        accx = fmaf(v2, a2.x, accx); accy = fmaf(v2, a2.y, accy);
        accx = fmaf(v3, a3.x, accx); accy = fmaf(v3, a3.y, accy);
    }
    for (; e < end; ++e) {
        const int c = (int)indices[e];
        const float v = values[e];
        const float2 a = x2[c * (DFEAT / 2) + lane];
        accx = fmaf(v, a.x, accx);
        accy = fmaf(v, a.y, accy);
    }
#endif

    float2* out2 = reinterpret_cast<float2*>(out);
    out2[(long long)row * (DFEAT / 2) + lane] = make_float2(accx, accy);
}

extern "C" void kernel_launch(void* const* d_in, const int* in_sizes, int n_in,
                              void* d_out, int out_size, void* d_ws, size_t ws_size,
                              hipStream_t stream) {
    // setup_inputs order: x (f32), indptr (i64), indices (i64), values (f32)
    const float*     x       = (const float*)d_in[0];
    const long long* indptr  = (const long long*)d_in[1];
    const long long* indices = (const long long*)d_in[2];
    const float*     values  = (const float*)d_in[3];
    float*           out     = (float*)d_out;

    const int n_rows = in_sizes[1] - 1;   // indptr has N+1 entries
    const int blocks = (n_rows + WAVES_PER_BLOCK - 1) / WAVES_PER_BLOCK;

    spmm_csr_tdm<<<blocks, THREADS_PER_BLOCK, 0, stream>>>(
        x, indptr, indices, values, out, n_rows);
}